// WeakNegativeLoss_89807766159905
// MI455X (gfx1250) — compile-verified
//
#include <hip/hip_runtime.h>
#include <math.h>

#define THETA 0.1f
#define EPSF  1e-7f
#define BDIM  32
#define LLEN  2048
#define DIM   128
#define KEYS  64
#define DSTRIDE 132   // 128 + 4 pad: distinct banks across 16 rows, 16B-aligned rows

typedef __attribute__((ext_vector_type(2))) float v2f;
typedef __attribute__((ext_vector_type(8))) float v8f;

// ws layout (floats): [0]=sum(pos*log(p+eps)); [1..32]=wnl max per b (uint bits); [33..64]=neg row sum per b
#define WS_POS 0
#define WS_WNL 1
#define WS_NEG 33
#define WS_N   65

__global__ void init_ws(float* ws) {
    int t = threadIdx.x;
    if (t < WS_N) ws[t] = 0.0f;
}

__global__ __launch_bounds__(128) void wnl_kernel(const float* __restrict__ emb,
                                                  const int* __restrict__ labels,
                                                  float* __restrict__ ws) {
    __shared__ float sk[KEYS * DSTRIDE];  // staged keys
    __shared__ float sq[DIM];             // raw query e[b,b,:]
    __shared__ float sinv[KEYS];          // 1/||key||
    __shared__ float spos[KEYS];          // pos labels as float
    __shared__ float sinvq;               // 1/||q||

    const int b   = blockIdx.y;
    const int k0  = blockIdx.x * KEYS;
    const int tid = threadIdx.x;
    const int w   = tid >> 5;
    const int ln  = tid & 31;

    const float* ebase = emb + ((size_t)b * LLEN + k0) * DIM;

    // ---- stage 64 keys x 128 dims via async global->LDS B128 copies ----
    // Per lane: 16 B straight to LDS, no VGPR staging; tracked by ASYNCcnt.
    #pragma unroll
    for (int i = 0; i < 16; ++i) {
        int idx = tid + i * 128;          // 0..2047 16B slots
        int row = idx >> 5;               // key row 0..63
        int c4  = idx & 31;               // float4 column
        unsigned ldsa = (unsigned)(size_t)&sk[row * DSTRIDE + c4 * 4]; // LDS byte offset
        unsigned goff = (unsigned)((row * DIM + c4 * 4) * 4);          // global byte offset
        asm volatile("global_load_async_to_lds_b128 %0, %1, %2"
                     :: "v"(ldsa), "v"(goff), "s"(ebase)
                     : "memory");
    }

    if (w == 0) {
        // query vector + its inverse norm (overlaps the async copies)
        const float* qg = emb + ((size_t)b * LLEN + b) * DIM;
        float4 v = *(const float4*)(qg + ln * 4);
        *(float4*)(&sq[ln * 4]) = v;
        float s = v.x*v.x + v.y*v.y + v.z*v.z + v.w*v.w;
        for (int off = 16; off > 0; off >>= 1) s += __shfl_xor(s, off);
        if (ln == 0) sinvq = 1.0f / fmaxf(sqrtf(s), 1e-12f);
    } else if (w == 1) {
        spos[ln]      = (float)labels[(size_t)b * LLEN + k0 + ln];
        spos[ln + 32] = (float)labels[(size_t)b * LLEN + k0 + ln + 32];
    }

    // drain this wave's async copies, then block-wide barrier
    asm volatile("s_wait_asynccnt 0x0" ::: "memory");
    __syncthreads();

    // ---- per-key inverse norms from LDS (2 lanes per key, 64 dims each) ----
    {
        int key = w * 16 + (ln >> 1);
        const float* r = &sk[key * DSTRIDE + (ln & 1) * 64];
        float s = 0.0f;
        #pragma unroll
        for (int d = 0; d < 64; ++d) { float x = r[d]; s += x * x; }
        s += __shfl_xor(s, 1);
        if ((ln & 1) == 0) sinv[key] = 1.0f / fmaxf(sqrtf(s), 1e-12f);
    }
    __syncthreads();

    // ---- fp32 WMMA: dots of query vs this wave's 16 keys; K=4 per op, 32 ops over D=128 ----
    // A (16x4 f32): lane ln -> M = ln%16, VGPR j -> K = j + 2*(ln>=16)
    // B (4x16 f32): all 16 columns = query chunk (same per-lane K striping)
    const int mrow = w * 16 + (ln & 15);
    const int kk   = (ln >> 4) * 2;
    const float* arow = &sk[mrow * DSTRIDE];
    v8f acc = {};
    #pragma unroll 4
    for (int d0 = 0; d0 < DIM; d0 += 4) {
        v2f a, bq;
        a.x  = arow[d0 + kk];
        a.y  = arow[d0 + kk + 1];
        bq.x = sq[d0 + kk];
        bq.y = sq[d0 + kk + 1];
        acc = __builtin_amdgcn_wmma_f32_16x16x4_f32(false, a, false, bq,
                                                    (short)0, acc, false, false);
    }

    // ---- extract: C/D layout -> VGPR i holds key (w*16 + i + 8*(ln>=16)); all columns equal ----
    const float invq = sinvq;
    const int hl = ln >> 4;
    float m = 0.0f;
    #pragma unroll
    for (int i = 0; i < 8; ++i) {
        int key = w * 16 + hl * 8 + i;
        float cosv = acc[i] * invq * sinv[key];
        float beta = fmaxf(cosv - THETA, 0.0f);
        m = fmaxf(m, beta * spos[key]);
    }
    for (int off = 16; off > 0; off >>= 1) m = fmaxf(m, __shfl_xor(m, off));
    if (ln == 0)
        atomicMax((unsigned int*)&ws[WS_WNL + b], __float_as_uint(m));
}

__global__ __launch_bounds__(256) void loss_kernel(const float* __restrict__ probs,
                                                   const int* __restrict__ labels,
                                                   float* __restrict__ ws) {
    const int b = blockIdx.x;
    const int t = threadIdx.x;
    float pl = 0.0f, nl = 0.0f;
    #pragma unroll
    for (int j = 0; j < LLEN / 256; ++j) {
        int idx = t + j * 256;
        float p = probs[(size_t)b * LLEN + idx];
        int lab = labels[(size_t)b * LLEN + idx];
        pl += (float)lab * logf(p + EPSF);
        float x = 1.0f - (p + EPSF);
        x = fminf(fmaxf(x, 1e-7f), 1.0f - 1e-7f);
        nl += logf(x);
    }
    __shared__ float rp[8], rn[8];
    int w = t >> 5, ln = t & 31;
    for (int off = 16; off > 0; off >>= 1) {
        pl += __shfl_xor(pl, off);
        nl += __shfl_xor(nl, off);
    }
    if (ln == 0) { rp[w] = pl; rn[w] = nl; }
    __syncthreads();
    if (t == 0) {
        float sp = 0.0f, sn = 0.0f;
        #pragma unroll
        for (int i = 0; i < 8; ++i) { sp += rp[i]; sn += rn[i]; }
        atomicAdd(&ws[WS_POS], sp);
        ws[WS_NEG + b] = sn;   // one block per b: plain store
    }
}

__global__ void final_kernel(const int* __restrict__ labels,
                             const float* __restrict__ ws,
                             float* __restrict__ out) {
    int ln = threadIdx.x;
    float part = 0.0f;
    if (ln < BDIM) {
        float wnl = __uint_as_float(((const unsigned int*)ws)[WS_WNL + ln]);
        int lab = labels[(size_t)ln * LLEN + ln];
        float diag = (lab == 0) ? wnl : 0.0f;
        part = diag * ws[WS_NEG + ln];
    }
    for (int off = 16; off > 0; off >>= 1) part += __shfl_xor(part, off);
    if (ln == 0) {
        float posloss = -ws[WS_POS];
        float negloss = -part;
        out[0] = (posloss + negloss) / (float)BDIM;
    }
}

extern "C" void kernel_launch(void* const* d_in, const int* in_sizes, int n_in,
                              void* d_out, int out_size, void* d_ws, size_t ws_size,
                              hipStream_t stream) {
    const float* emb    = (const float*)d_in[0];  // (32, 2048, 128) f32
    const float* probs  = (const float*)d_in[1];  // (32, 2048) f32
    const int*   labels = (const int*)d_in[2];    // (32, 2048) i32
    float* out = (float*)d_out;
    float* ws  = (float*)d_ws;

    hipLaunchKernelGGL(init_ws,      dim3(1),                 dim3(128), 0, stream, ws);
    hipLaunchKernelGGL(wnl_kernel,   dim3(LLEN / KEYS, BDIM), dim3(128), 0, stream, emb, labels, ws);
    hipLaunchKernelGGL(loss_kernel,  dim3(BDIM),              dim3(256), 0, stream, probs, labels, ws);
    hipLaunchKernelGGL(final_kernel, dim3(1),                 dim3(32),  0, stream, labels, ws, out);
}